// DarkChannelLoss_55748675502138
// MI455X (gfx1250) — compile-verified
//
#include <hip/hip_runtime.h>
#include <stdint.h>

// ---------------------------------------------------------------------------
// DarkChannelLoss for MI455X (gfx1250)
//   dc = sliding 15x15 clipped min over reflect-padded channel-min image
//   out = -mean(dc)  (scalar)
// Fused single-pass tile kernel: async global->LDS loads (ASYNCcnt path),
// separable log-step sliding min in LDS, WMMA f32 16x16x4 wave reduction.
// ---------------------------------------------------------------------------

#define WIN   15
#define PADW  7
#define NPIX  512
#define HP    526           // NPIX + 2*PADW
#define TILE  64
#define REGN  78            // TILE + WIN - 1
#define LSTR  80            // LDS row stride in floats
#define NTILE 9             // ceil(HP / TILE)
#define BATCH 32
#define NBLK  (NTILE * NTILE * BATCH)

typedef __attribute__((ext_vector_type(2))) float v2f;
typedef __attribute__((ext_vector_type(8))) float v8f;

__device__ __forceinline__ int reflect7(int g) {
  // padded index g in [0, HP) -> source index in [0, NPIX)
  int s = g - PADW;
  s = (s < 0) ? -s : s;
  s = (s > NPIX - 1) ? (2 * (NPIX - 1) - s) : s;
  return s;
}

// CDNA5 async global -> LDS copy (4 bytes per active lane), ASYNCcnt-tracked.
__device__ __forceinline__ void async_ld_b32(uint32_t lds_off, const float* gptr) {
  asm volatile("global_load_async_to_lds_b32 %0, %1, off"
               :: "v"(lds_off), "v"(gptr)
               : "memory");
}
__device__ __forceinline__ void wait_async0() {
  asm volatile("s_wait_asynccnt 0" ::: "memory");
}

__global__ __launch_bounds__(256)
void dark_tile_kernel(const float* __restrict__ img, float* __restrict__ partial) {
  __shared__ float r0[REGN * LSTR];
  __shared__ float r1[REGN * LSTR];
  __shared__ float wsum[8];

  const int tid = threadIdx.x;
  const int i0  = blockIdx.y * TILE;
  const int j0  = blockIdx.x * TILE;
  const int b   = blockIdx.z;
  const float INF = __builtin_inff();

  const float* p0 = img + ((size_t)b * 3 + 0) * (size_t)(NPIX * NPIX);
  const float* p1 = p0 + (size_t)(NPIX * NPIX);
  const float* p2 = p1 + (size_t)(NPIX * NPIX);

  const uint32_t l0 = (uint32_t)(size_t)(void*)r0;
  const uint32_t l1 = (uint32_t)(size_t)(void*)r1;

  // ---- load channels 0,1 of the 78x78 halo region (reflect-indexed) -------
  for (int r = (tid >> 6); r < REGN; r += 4) {
    const int gr = i0 + r;
    const int sr = reflect7((gr < HP) ? gr : 0);
    const size_t rowoff = (size_t)sr * NPIX;
    for (int c = (tid & 63); c < REGN; c += 64) {
      const int gc  = j0 + c;
      const int idx = r * LSTR + c;
      if (gr < HP && gc < HP) {
        const int sc = reflect7(gc);
        async_ld_b32(l0 + 4u * (uint32_t)idx, p0 + rowoff + sc);
        async_ld_b32(l1 + 4u * (uint32_t)idx, p1 + rowoff + sc);
      } else {
        r0[idx] = INF;   // clipped-window semantics: +inf outside padded grid
        r1[idx] = INF;
      }
    }
  }
  wait_async0();
  __syncthreads();

  // r0 = min(ch0, ch1)
  for (int r = (tid >> 6); r < REGN; r += 4)
    for (int c = (tid & 63); c < REGN; c += 64) {
      const int idx = r * LSTR + c;
      r0[idx] = fminf(r0[idx], r1[idx]);
    }
  __syncthreads();

  // ---- load channel 2 into r1, fold in ------------------------------------
  for (int r = (tid >> 6); r < REGN; r += 4) {
    const int gr = i0 + r;
    const int sr = reflect7((gr < HP) ? gr : 0);
    const size_t rowoff = (size_t)sr * NPIX;
    for (int c = (tid & 63); c < REGN; c += 64) {
      const int gc  = j0 + c;
      const int idx = r * LSTR + c;
      if (gr < HP && gc < HP) {
        async_ld_b32(l1 + 4u * (uint32_t)idx, p2 + rowoff + reflect7(gc));
      } else {
        r1[idx] = INF;
      }
    }
  }
  wait_async0();
  __syncthreads();

  for (int r = (tid >> 6); r < REGN; r += 4)
    for (int c = (tid & 63); c < REGN; c += 64) {
      const int idx = r * LSTR + c;
      r0[idx] = fminf(r0[idx], r1[idx]);   // m = channel min
    }
  __syncthreads();

  // ---- separable sliding min, log-step decomposition ----------------------
  // col pass: out valid cols 0..63, rows 0..77
  auto cs = [&](float* dst, const float* src, int ncols, int d) {
    for (int r = (tid >> 6); r < REGN; r += 4)
      for (int c = (tid & 63); c < ncols; c += 64)
        dst[r * LSTR + c] = fminf(src[r * LSTR + c], src[r * LSTR + c + d]);
    __syncthreads();
  };
  cs(r1, r0, 77, 1);   // width 2
  cs(r0, r1, 75, 2);   // width 4
  cs(r1, r0, 71, 4);   // width 8
  cs(r0, r1, 64, 7);   // width 15 = min(w8[j], w8[j+7])

  // row pass: out valid rows 0..63, cols 0..63
  auto rs = [&](float* dst, const float* src, int nrows, int d) {
    const int c = tid & 63;
    for (int r = (tid >> 6); r < nrows; r += 4)
      dst[r * LSTR + c] = fminf(src[r * LSTR + c], src[(r + d) * LSTR + c]);
    __syncthreads();
  };
  rs(r1, r0, 77, 1);
  rs(r0, r1, 75, 2);
  rs(r1, r0, 71, 4);   // r1 = width-8 row min

  // ---- final width-15 row combine + masked accumulation -------------------
  float acc0 = 0.0f, acc1 = 0.0f;
  {
    const int c = tid & 63;
    const bool colok = (j0 + c) < HP;
    int k = 0;
    for (int r = (tid >> 6); r < TILE; r += 4, ++k) {
      const float v = fminf(r1[r * LSTR + c], r1[(r + 7) * LSTR + c]);
      const float add = (colok && (i0 + r) < HP) ? v : 0.0f;
      if (k & 1) acc1 += add; else acc0 += add;
    }
  }

  // ---- wave reduction via WMMA: D = A(16x4 accs) x B(4x16 ones) -----------
  v2f a;   a.x = acc0; a.y = acc1;
  v2f one; one.x = 1.0f; one.y = 1.0f;
  v8f cz = {};
  v8f d = __builtin_amdgcn_wmma_f32_16x16x4_f32(
      /*neg_a=*/false, a, /*neg_b=*/false, one,
      /*c_mod=*/(short)0, cz, /*reuse_a=*/false, /*reuse_b=*/false);
  // lane 0 holds D[0..7][0], lane 16 holds D[8..15][0] -> column-0 sum = wave total
  float t = d[0] + d[1] + d[2] + d[3] + d[4] + d[5] + d[6] + d[7];
  const float wtot = __shfl(t, 0, 32) + __shfl(t, 16, 32);

  const int lane = tid & 31, wid = tid >> 5;
  if (lane == 0) wsum[wid] = wtot;
  __syncthreads();
  if (tid == 0) {
    float s = 0.0f;
    for (int w = 0; w < 8; ++w) s += wsum[w];
    partial[((size_t)blockIdx.z * NTILE + blockIdx.y) * NTILE + blockIdx.x] = s;
  }
}

// Deterministic fixed-order final reduction (no float atomics anywhere).
__global__ __launch_bounds__(256)
void dark_reduce_kernel(const float* __restrict__ partial, float* __restrict__ out) {
  __shared__ float s[256];
  float a = 0.0f;
  for (int i = threadIdx.x; i < NBLK; i += 256) a += partial[i];
  s[threadIdx.x] = a;
  __syncthreads();
  for (int step = 128; step > 0; step >>= 1) {
    if (threadIdx.x < step) s[threadIdx.x] += s[threadIdx.x + step];
    __syncthreads();
  }
  if (threadIdx.x == 0) {
    const float scale = 1.0f / ((float)BATCH * (float)HP * (float)HP);
    out[0] = -s[0] * scale;
  }
}

extern "C" void kernel_launch(void* const* d_in, const int* in_sizes, int n_in,
                              void* d_out, int out_size, void* d_ws, size_t ws_size,
                              hipStream_t stream) {
  (void)in_sizes; (void)n_in; (void)out_size; (void)ws_size;
  const float* img = (const float*)d_in[0];
  float* partial = (float*)d_ws;      // NBLK floats of scratch
  float* out = (float*)d_out;

  dim3 grid(NTILE, NTILE, BATCH);
  dark_tile_kernel<<<grid, 256, 0, stream>>>(img, partial);
  dark_reduce_kernel<<<1, 256, 0, stream>>>(partial, out);
}